// FM_81140522156060
// MI455X (gfx1250) — compile-verified
//
#include <hip/hip_runtime.h>

typedef float v2f __attribute__((ext_vector_type(2)));
typedef float v8f __attribute__((ext_vector_type(8)));

#define FEAS 4096
#define NROWS 4096
#define KC 64          // cross inner dim
#define CROSS_BLOCKS 64
#define PART_STRIDE 65 // 64 col partials + 1 diag partial per block
#define KWAVES 8       // K-dim split per 16-row tile
#define KSPAN (FEAS / KWAVES) // 512 columns per wave

// ---------------------------------------------------------------------------
// Kernel 1: per-block partial column sums + partial sum-of-squares of V(4096x64)
// Deterministic: fixed-order LDS tree reductions, no atomics.
// ---------------------------------------------------------------------------
__global__ __launch_bounds__(256) void fm_cross_partial(const float* __restrict__ V,
                                                        float* __restrict__ part) {
    __shared__ float s_col[256];
    __shared__ float s_sq[256];
    const int tid  = threadIdx.x;
    const int gtid = blockIdx.x * 256 + tid;
    const int stride = 256 * CROSS_BLOCKS; // 16384, multiple of 64 -> column fixed per thread
    float s = 0.0f, q = 0.0f;
    for (int i = gtid; i < FEAS * KC; i += stride) {
        float v = V[i];
        s += v;
        q += v * v;
    }
    s_col[tid] = s;
    s_sq[tid]  = q;
    __syncthreads();
    // threads tid, tid+64, tid+128, tid+192 all hold the same column (tid % 64)
    if (tid < 64) {
        float c = s_col[tid] + s_col[tid + 64] + s_col[tid + 128] + s_col[tid + 192];
        part[blockIdx.x * PART_STRIDE + tid] = c;
    }
    for (int off = 128; off > 0; off >>= 1) {
        if (tid < off) s_sq[tid] += s_sq[tid + off];
        __syncthreads();
    }
    if (tid == 0) part[blockIdx.x * PART_STRIDE + 64] = s_sq[0];
}

// ---------------------------------------------------------------------------
// Kernel 2: fold partials -> cross_sum scalar, fuse +linear_b. One tiny block.
// ---------------------------------------------------------------------------
__global__ __launch_bounds__(64) void fm_cross_final(const float* __restrict__ part,
                                                     const float* __restrict__ linear_b,
                                                     float* __restrict__ scalar_out) {
    __shared__ float s_cs2[64];
    const int tid = threadIdx.x;
    float c = 0.0f;
    for (int b = 0; b < CROSS_BLOCKS; ++b) c += part[b * PART_STRIDE + tid];
    s_cs2[tid] = c * c;
    __syncthreads();
    if (tid == 0) {
        float total = 0.0f;
        for (int j = 0; j < 64; ++j) total += s_cs2[j];
        float diag = 0.0f;
        for (int b = 0; b < CROSS_BLOCKS; ++b) diag += part[b * PART_STRIDE + 64];
        scalar_out[0] = 0.5f * (total + diag) + linear_b[0];
    }
}

// ---------------------------------------------------------------------------
// Kernel 3: linear = x @ w via V_WMMA_F32_16X16X4_F32, B broadcast across N.
// Block = 256 threads (8 waves). Each block owns one 16-row tile; each wave
// covers 512 of the 4096 K-columns (2048 waves total -> enough loads in
// flight to cover HBM latency). x streams through VMEM (global_load_b64 per
// the 32-bit A 16x4 lane layout); w comes via the scalar path (uniform) and
// is expanded to the B fragment with cndmask selects, so the vector memory
// pipe is dedicated to x.
// ---------------------------------------------------------------------------
__global__ __launch_bounds__(256) void fm_linear_wmma(const float* __restrict__ x,
                                                      const float* __restrict__ w,
                                                      const float* __restrict__ shift,
                                                      float* __restrict__ out) {
    __shared__ float s_part[KWAVES][16];
    const int tid  = threadIdx.x;
    const int wave = __builtin_amdgcn_readfirstlane(tid >> 5); // force SGPR
    const int lane = tid & 31;
    const int hi   = lane >> 4;   // half-wave (K-pair selector)
    const bool hiB = (hi != 0);
    const int m    = lane & 15;   // row within tile
    const int rowBase = blockIdx.x * 16;
    const int k0   = wave * KSPAN;

    const float* __restrict__ xp = x + (size_t)(rowBase + m) * FEAS + k0 + 2 * hi;
    const float* __restrict__ wp = w + k0;   // uniform address -> s_load

    v8f acc = {};
    #pragma unroll 16
    for (int kk = 0; kk < KSPAN; kk += 4) {
        // A fragment: lane holds x[row][k0+kk + 2*hi .. +1] (ISA 32-bit A 16x4 layout)
        v2f a = *(const v2f*)(xp + kk);
        // B fragment: w broadcast across all 16 columns; scalar loads + selects
        float w0 = wp[kk + 0], w1 = wp[kk + 1], w2 = wp[kk + 2], w3 = wp[kk + 3];
        v2f b;
        b.x = hiB ? w2 : w0;
        b.y = hiB ? w3 : w1;
        acc = __builtin_amdgcn_wmma_f32_16x16x4_f32(
            /*neg_a=*/false, a, /*neg_b=*/false, b,
            /*c_mod=*/(short)0, acc, /*reuse_a=*/false, /*reuse_b=*/false);
    }

    // C/D layout: VGPR r -> (M = r + 8*hi, N = m). Columns identical, so lanes
    // 0 and 16 carry all 16 row partials between them.
    if (m == 0) {
        #pragma unroll
        for (int r = 0; r < 8; ++r) s_part[wave][8 * hi + r] = acc[r];
    }
    __syncthreads();

    if (tid < 16) {
        float v = 0.0f;
        #pragma unroll
        for (int w8 = 0; w8 < KWAVES; ++w8) v += s_part[w8][tid];
        v += shift[0];                                   // cross_sum + linear_b
        out[rowBase + tid] = 1.0f / (1.0f + __expf(-v)); // sigmoid
    }
}

// ---------------------------------------------------------------------------
extern "C" void kernel_launch(void* const* d_in, const int* in_sizes, int n_in,
                              void* d_out, int out_size, void* d_ws, size_t ws_size,
                              hipStream_t stream) {
    const float* x     = (const float*)d_in[0]; // (4096, 4096)
    const float* cross = (const float*)d_in[1]; // (4096, 1, 64)
    const float* w     = (const float*)d_in[2]; // (1, 4096)
    const float* b     = (const float*)d_in[3]; // (1,)
    float* out = (float*)d_out;                 // (4096, 1)

    float* part   = (float*)d_ws;               // 64 * 65 floats
    float* scalar = part + CROSS_BLOCKS * PART_STRIDE;

    fm_cross_partial<<<CROSS_BLOCKS, 256, 0, stream>>>(cross, part);
    fm_cross_final<<<1, 64, 0, stream>>>(part, b, scalar);
    fm_linear_wmma<<<NROWS / 16, 256, 0, stream>>>(x, w, scalar, out);
}